// Net_75084618268982
// MI455X (gfx1250) — compile-verified
//
#include <hip/hip_runtime.h>

// ---------------------------------------------------------------------------
// SNN scan (T=50) for MI455X / gfx1250.
// Three big GEMMs per step run on V_WMMA_F32_16X16X4_F32 (the only fp32 WMMA
// shape) to preserve fp32 recurrence semantics. wave32, one wave per tile.
// ---------------------------------------------------------------------------

typedef __attribute__((ext_vector_type(2))) float v2f;
typedef __attribute__((ext_vector_type(8))) float v8f;

constexpr int T_ = 50, NB = 256, NI = 784, NH = 2048, NO = 10;
constexpr float BETA_   = 0.9f;
constexpr float DECAY_  = 1.0f / 20.0f;   // DT / STDP_DECAY
constexpr float APLUS_  = 0.005f;
constexpr float AMINUS_ = 0.005f;
constexpr float INVB_   = 1.0f / (float)NB;

__device__ __forceinline__ v8f wmma_f32(v2f a, v2f b, v8f c) {
  // D = A(16x4) * B(4x16) + C(16x16), all fp32
  return __builtin_amdgcn_wmma_f32_16x16x4_f32(
      /*neg_a=*/false, a, /*neg_b=*/false, b,
      /*c_mod=*/(short)0, c, /*reuse_a=*/false, /*reuse_b=*/false);
}

// LIF (reset_mechanism='zero', thr=1) + STDP trace update + record stores.
__device__ __forceinline__ void lif_zero_trace_store(
    float cur, size_t idx,
    float* __restrict__ mem, float* __restrict__ pre, float* __restrict__ post,
    float* __restrict__ spkws,
    float* __restrict__ out_spk, float* __restrict__ out_pre,
    float* __restrict__ out_post, float* __restrict__ out_mem) {
  float m = mem[idx];
  float reset = (m - 1.0f > 0.0f) ? 1.0f : 0.0f;
  float mn = (BETA_ * m + cur) * (1.0f - reset);
  float s  = (mn - 1.0f > 0.0f) ? 1.0f : 0.0f;
  mem[idx] = mn;
  float p = pre[idx];  p = p - DECAY_ * p + APLUS_ * s;  pre[idx] = p;
  float q = post[idx]; q = q - DECAY_ * q - AMINUS_ * s; post[idx] = q;
  if (spkws)   spkws[idx]  = s;
  out_spk[idx]  = s;
  out_pre[idx]  = p;
  out_post[idx] = q;
  if (out_mem) out_mem[idx] = mn;
}

// ---------------------------------------------------------------------------
// Kernel 0: zero-init workspace state (grid-stride).
// ---------------------------------------------------------------------------
__global__ void k_zero(float* __restrict__ p, size_t n) {
  size_t i = (size_t)blockIdx.x * blockDim.x + threadIdx.x;
  size_t stride = (size_t)gridDim.x * blockDim.x;
  for (; i < n; i += stride) p[i] = 0.0f;
}

// ---------------------------------------------------------------------------
// Kernel 1: LIF layer 0 (subtract-reset, thr=0 -> no subtraction) + trace0.
// One thread per (b,i); B*I = 200704 elements.
// ---------------------------------------------------------------------------
__global__ void k_lif0(const float* __restrict__ xt,
                       float* __restrict__ mem0, float* __restrict__ pre0,
                       float* __restrict__ post0, float* __restrict__ spk0ws,
                       float* __restrict__ out_spk0, float* __restrict__ out_pre0,
                       float* __restrict__ out_post0) {
  size_t idx = (size_t)blockIdx.x * blockDim.x + threadIdx.x;   // < NB*NI
  float m  = mem0[idx];
  float mn = BETA_ * m + xt[idx];                 // thr==0: reset term vanishes
  float s  = (mn > 0.0f) ? 1.0f : 0.0f;
  mem0[idx] = mn;
  float p = pre0[idx];  p = p - DECAY_ * p + APLUS_ * s;  pre0[idx] = p;
  float q = post0[idx]; q = q - DECAY_ * q - AMINUS_ * s; post0[idx] = q;
  spk0ws[idx]   = s;
  out_spk0[idx] = s;
  out_pre0[idx] = p;
  out_post0[idx] = q;
}

// ---------------------------------------------------------------------------
// Kernel 2: cur1 = relu(spk0 @ w1^T) via fp32 WMMA, fused LIF1 + trace1.
// One wave per 16(batch) x 32(hidden) output tile: two accumulators share the
// A fragment. K = NI = 784 = outer loop of 49 x (prefetch + 4 branch-free
// WMMA steps). Speculative prefetch past panel end is dropped by HW.
//   A[m][k] = spk0[b0+m][k]          (A fragment: lane<16 -> K={k,k+1},
//                                     lane>=16 -> K={k+2,k+3}, M=lane&15)
//   B[k][n] = w1[h0+n][k]
// C layout: VGPR v -> M = v + 8*(lane>>4), N = lane&15.
// ---------------------------------------------------------------------------
__global__ __launch_bounds__(32) void k_gemm1_lif1(
    const float* __restrict__ spk0, const float* __restrict__ w1c,
    float* __restrict__ mem1, float* __restrict__ pre1,
    float* __restrict__ post1, float* __restrict__ spk1ws,
    float* __restrict__ out_spk1, float* __restrict__ out_pre1,
    float* __restrict__ out_post1) {
  const int lane  = threadIdx.x;
  const int row   = lane & 15;
  const int khalf = lane >> 4;           // 0 or 1
  const int h0 = blockIdx.x * 32;
  const int bt0 = blockIdx.y * 16;

  const float* __restrict__ arow  = spk0 + (size_t)(bt0 + row) * NI + 2 * khalf;
  const float* __restrict__ brow0 = w1c  + (size_t)(h0 + row) * NI + 2 * khalf;
  const float* __restrict__ brow1 = w1c  + (size_t)(h0 + 16 + row) * NI + 2 * khalf;

  v8f acc0 = {}, acc1 = {};
  for (int k0 = 0; k0 < NI; k0 += 16) {
    // One prefetch window ahead; branch-free (speculative, OOB is dropped).
    __builtin_prefetch(arow + k0 + 64, 0, 3);
    __builtin_prefetch(brow0 + k0 + 64, 0, 3);
    __builtin_prefetch(brow1 + k0 + 64, 0, 3);
#pragma unroll
    for (int k = k0; k < k0 + 16; k += 4) {
      v2f a   = { arow[k],  arow[k + 1] };
      v2f bb0 = { brow0[k], brow0[k + 1] };
      v2f bb1 = { brow1[k], brow1[k + 1] };
      acc0 = wmma_f32(a, bb0, acc0);
      acc1 = wmma_f32(a, bb1, acc1);
    }
  }

#pragma unroll
  for (int v = 0; v < 8; ++v) {
    int b = bt0 + v + 8 * khalf;
    {
      size_t idx = (size_t)b * NH + (h0 + row);
      lif_zero_trace_store(fmaxf(acc0[v], 0.0f), idx, mem1, pre1, post1, spk1ws,
                           out_spk1, out_pre1, out_post1, nullptr);
    }
    {
      size_t idx = (size_t)b * NH + (h0 + 16 + row);
      lif_zero_trace_store(fmaxf(acc1[v], 0.0f), idx, mem1, pre1, post1, spk1ws,
                           out_spk1, out_pre1, out_post1, nullptr);
    }
  }
}

// ---------------------------------------------------------------------------
// Kernel 3: cur2 = relu(spk1 @ w2^T) (O=10) + LIF2 + trace2.
// One block per batch element: 256 threads, each covering 8 contiguous h
// (coalesced float4 reads of spk1 and w2), then an LDS tree reduction of the
// 10 per-output partials; threads 0..9 run the LIF/trace epilogue.
// ---------------------------------------------------------------------------
__global__ __launch_bounds__(256) void k_gemm2_lif2(
    const float* __restrict__ spk1ws, const float* __restrict__ w2,
    float* __restrict__ mem2, float* __restrict__ pre2,
    float* __restrict__ post2,
    float* __restrict__ out_spk2, float* __restrict__ out_pre2,
    float* __restrict__ out_post2, float* __restrict__ out_mem2) {
  __shared__ float red[256 * NO];
  const int b = blockIdx.x;
  const int t = threadIdx.x;

  const float4* __restrict__ s1 =
      reinterpret_cast<const float4*>(spk1ws + (size_t)b * NH) + t * 2;
  float4 a0 = s1[0], a1 = s1[1];

#pragma unroll
  for (int o = 0; o < NO; ++o) {
    const float4* __restrict__ wr =
        reinterpret_cast<const float4*>(w2 + (size_t)o * NH) + t * 2;
    float4 w0 = wr[0], w1v = wr[1];
    red[t * NO + o] = a0.x * w0.x + a0.y * w0.y + a0.z * w0.z + a0.w * w0.w
                    + a1.x * w1v.x + a1.y * w1v.y + a1.z * w1v.z + a1.w * w1v.w;
  }
  __syncthreads();
#pragma unroll
  for (int s = 128; s > 0; s >>= 1) {
    if (t < s) {
#pragma unroll
      for (int o = 0; o < NO; ++o) red[t * NO + o] += red[(t + s) * NO + o];
    }
    __syncthreads();
  }
  if (t < NO) {
    size_t idx = (size_t)b * NO + t;
    lif_zero_trace_store(fmaxf(red[t], 0.0f), idx, mem2, pre2, post2, nullptr,
                         out_spk2, out_pre2, out_post2, out_mem2);
  }
}

// ---------------------------------------------------------------------------
// Kernel 4: fused STDP update of w1.
//   adjA = post1^T @ spk0 / B   (M=H, N=I, K=B)
//   adjB = spk1^T  @ pre0 / B
//   w = clip(w + |w|*adjA); w = clip(w + |w|*adjB)
// One wave per 16x16 tile of [H,I]; K = NB = 256 = outer loop of 16 x
// (prefetch + 4 branch-free steps, 2 WMMA each). Fragment loads are
// coalesced: each half-wave reads 64B-contiguous row segments.
// ---------------------------------------------------------------------------
__global__ __launch_bounds__(32) void k_stdp(
    const float* __restrict__ post1, const float* __restrict__ spk0ws,
    const float* __restrict__ spk1ws, const float* __restrict__ pre0,
    float* __restrict__ w1c) {
  const int lane  = threadIdx.x;
  const int row   = lane & 15;
  const int khalf = lane >> 4;
  const int i0 = blockIdx.x * 16;
  const int h0 = blockIdx.y * 16;

  const float* __restrict__ p1c = post1  + (h0 + row);   // A1[m][k] = post1[k][h0+m]
  const float* __restrict__ s1c = spk1ws + (h0 + row);   // A2[m][k] = spk1[k][h0+m]
  const float* __restrict__ s0c = spk0ws + (i0 + row);   // B1[k][n] = spk0[k][i0+n]
  const float* __restrict__ p0c = pre0   + (i0 + row);   // B2[k][n] = pre0[k][i0+n]

  v8f accA = {}, accB = {};
  for (int k0 = 0; k0 < NB; k0 += 16) {
    __builtin_prefetch(p1c + (size_t)(k0 + 16) * NH, 0, 3);
    __builtin_prefetch(s1c + (size_t)(k0 + 16) * NH, 0, 3);
    __builtin_prefetch(s0c + (size_t)(k0 + 16) * NI, 0, 3);
    __builtin_prefetch(p0c + (size_t)(k0 + 16) * NI, 0, 3);
#pragma unroll
    for (int k = k0; k < k0 + 16; k += 4) {
      int kk = k + 2 * khalf;
      v2f a1 = { p1c[(size_t)kk * NH], p1c[(size_t)(kk + 1) * NH] };
      v2f b1 = { s0c[(size_t)kk * NI], s0c[(size_t)(kk + 1) * NI] };
      accA = wmma_f32(a1, b1, accA);
      v2f a2 = { s1c[(size_t)kk * NH], s1c[(size_t)(kk + 1) * NH] };
      v2f b2 = { p0c[(size_t)kk * NI], p0c[(size_t)(kk + 1) * NI] };
      accB = wmma_f32(a2, b2, accB);
    }
  }

#pragma unroll
  for (int v = 0; v < 8; ++v) {
    int h = h0 + v + 8 * khalf;
    int i = i0 + row;
    size_t wi = (size_t)h * NI + i;
    float w = w1c[wi];
    w = w + fabsf(w) * (accA[v] * INVB_);
    w = fminf(fmaxf(w, -1.0f), 1.0f);
    w = w + fabsf(w) * (accB[v] * INVB_);
    w = fminf(fmaxf(w, -1.0f), 1.0f);
    w1c[wi] = w;
  }
}

// ---------------------------------------------------------------------------
// Host side
// ---------------------------------------------------------------------------
extern "C" void kernel_launch(void* const* d_in, const int* in_sizes, int n_in,
                              void* d_out, int out_size, void* d_ws, size_t ws_size,
                              hipStream_t stream) {
  (void)in_sizes; (void)n_in; (void)out_size; (void)ws_size;
  const float* x  = (const float*)d_in[0];   // [T,B,I]
  const float* w1 = (const float*)d_in[1];   // [H,I]
  const float* w2 = (const float*)d_in[2];   // [O,H]
  float* out = (float*)d_out;
  float* ws  = (float*)d_ws;

  constexpr size_t BI = (size_t)NB * NI;     // 200704
  constexpr size_t BH = (size_t)NB * NH;     // 524288
  constexpr size_t BO = (size_t)NB * NO;     // 2560

  // Workspace layout (floats); state prefix is zero-initialized, w1c last.
  float* mem0  = ws;
  float* pre0  = mem0  + BI;
  float* post0 = pre0  + BI;
  float* spk0  = post0 + BI;
  float* mem1  = spk0  + BI;
  float* pre1  = mem1  + BH;
  float* post1 = pre1  + BH;
  float* spk1  = post1 + BH;
  float* mem2  = spk1  + BH;
  float* pre2  = mem2  + BO;
  float* post2 = pre2  + BO;
  float* w1c   = post2 + BO;
  const size_t state_count = 4 * BI + 4 * BH + 3 * BO;

  // Output record base offsets (return-order concatenation).
  constexpr size_t S_BI = (size_t)T_ * BI;
  constexpr size_t S_BH = (size_t)T_ * BH;
  constexpr size_t S_BO = (size_t)T_ * BO;
  float* o_spk0  = out;
  float* o_spk1  = o_spk0  + S_BI;
  float* o_spk2  = o_spk1  + S_BH;
  float* o_mem2  = o_spk2  + S_BO;
  float* o_pre0  = o_mem2  + S_BO;
  float* o_pre1  = o_pre0  + S_BI;
  float* o_pre2  = o_pre1  + S_BH;
  float* o_post0 = o_pre2  + S_BO;
  float* o_post1 = o_post0 + S_BI;
  float* o_post2 = o_post1 + S_BH;

  // Init: zero state, copy w1 into mutable workspace copy.
  k_zero<<<512, 256, 0, stream>>>(ws, state_count);
  hipMemcpyAsync(w1c, w1, (size_t)NH * NI * sizeof(float),
                 hipMemcpyDeviceToDevice, stream);

  for (int t = 0; t < T_; ++t) {
    const float* xt = x + (size_t)t * BI;
    k_lif0<<<(int)(BI / 256), 256, 0, stream>>>(
        xt, mem0, pre0, post0, spk0,
        o_spk0 + (size_t)t * BI, o_pre0 + (size_t)t * BI, o_post0 + (size_t)t * BI);

    k_gemm1_lif1<<<dim3(NH / 32, NB / 16), 32, 0, stream>>>(
        spk0, w1c, mem1, pre1, post1, spk1,
        o_spk1 + (size_t)t * BH, o_pre1 + (size_t)t * BH, o_post1 + (size_t)t * BH);

    k_gemm2_lif2<<<NB, 256, 0, stream>>>(
        spk1, w2, mem2, pre2, post2,
        o_spk2 + (size_t)t * BO, o_pre2 + (size_t)t * BO, o_post2 + (size_t)t * BO,
        o_mem2 + (size_t)t * BO);

    k_stdp<<<dim3(NI / 16, NH / 16), 32, 0, stream>>>(
        post1, spk0, spk1, pre0, w1c);
  }
}